// MoE_63342177681783
// MI455X (gfx1250) — compile-verified
//
#include <hip/hip_runtime.h>
#include <math.h>

typedef __attribute__((ext_vector_type(2))) float v2f;
typedef __attribute__((ext_vector_type(8))) float v8f;

#define B_ROWS 16384
#define D_DIM  4096
#define E_DIM  64
#define TOPK_N 8
#define KC     64   // K-chunk staged in LDS per iteration
#define M_TILE 64   // rows per workgroup (4 row-blocks of 16 per wave)

__global__ void moe_zero_load_kernel(float* __restrict__ load_out) {
    load_out[threadIdx.x] = 0.0f;
}

__global__ __launch_bounds__(256) void moe_router_kernel(
    const float* __restrict__ x,
    const float* __restrict__ w_gate,
    const float* __restrict__ w_noise,
    const float* __restrict__ noise,
    float* __restrict__ gates_out,
    float* __restrict__ load_out)
{
    __shared__ float As[M_TILE * KC];        // staged x tile (64 rows x KC)
    __shared__ float cleanL[M_TILE * 64];    // clean logits -> load contributions
    __shared__ float sdL[M_TILE * 64];       // pre-softplus noise logits -> stddev
    __shared__ float noisyL[M_TILE * 64];    // noisy logits
    __shared__ float probsL[M_TILE * 64];    // probs -> gates

    const int tid  = threadIdx.x;
    const int lane = tid & 31;
    const int wv   = tid >> 5;          // wave id 0..7
    const int m    = lane & 15;         // row/col within half-wave
    const int h    = lane >> 4;         // half-wave select
    const int row0 = blockIdx.x * M_TILE;

    // Waves 0-3: gate GEMM, waves 4-7: noise GEMM. Each owns four 16x16 tiles
    // stacked in M (rows rb*16 .. rb*16+15), sharing every B fragment 4 ways.
    const float* __restrict__ W = (wv < 4) ? w_gate : w_noise;
    const int n0 = (wv & 3) * 16;

    v8f acc0 = {0.f,0.f,0.f,0.f,0.f,0.f,0.f,0.f};
    v8f acc1 = acc0, acc2 = acc0, acc3 = acc0;

    // Staging assignment: thread -> 4 float4s (rows sr, sr+16, sr+32, sr+48)
    const int sr = tid >> 4;          // 0..15
    const int sc = (tid & 15) * 4;    // 0..60

    // Prefetch first chunk into registers
    float4 st0, st1, st2, st3;
    {
        const float* base = x + (size_t)row0 * D_DIM + sc;
        st0 = *(const float4*)(base + (size_t)(sr     ) * D_DIM);
        st1 = *(const float4*)(base + (size_t)(sr + 16) * D_DIM);
        st2 = *(const float4*)(base + (size_t)(sr + 32) * D_DIM);
        st3 = *(const float4*)(base + (size_t)(sr + 48) * D_DIM);
    }

    for (int kc = 0; kc < D_DIM; kc += KC) {
        __syncthreads();   // previous chunk fully consumed
        *(float4*)(&As[(sr     ) * KC + sc]) = st0;
        *(float4*)(&As[(sr + 16) * KC + sc]) = st1;
        *(float4*)(&As[(sr + 32) * KC + sc]) = st2;
        *(float4*)(&As[(sr + 48) * KC + sc]) = st3;
        __syncthreads();

        // Prefetch next chunk while this one computes
        if (kc + KC < D_DIM) {
            const float* base = x + (size_t)row0 * D_DIM + (kc + KC) + sc;
            st0 = *(const float4*)(base + (size_t)(sr     ) * D_DIM);
            st1 = *(const float4*)(base + (size_t)(sr + 16) * D_DIM);
            st2 = *(const float4*)(base + (size_t)(sr + 32) * D_DIM);
            st3 = *(const float4*)(base + (size_t)(sr + 48) * D_DIM);
        }

        #pragma unroll
        for (int ks = 0; ks < KC; ks += 4) {
            // B fragment (4x16): lanes 0-15 rows K 0/1; lanes 16-31 rows K 2/3
            v2f b;
            const float* wp = W + (size_t)(kc + ks + 2 * h) * E_DIM + n0 + m;
            b.x = wp[0];
            b.y = wp[E_DIM];
            // A fragments for the 4 row blocks
            const float* ap = &As[m * KC + ks + 2 * h];
            v2f a0, a1, a2, a3;
            a0.x = ap[0];           a0.y = ap[1];
            a1.x = ap[16 * KC];     a1.y = ap[16 * KC + 1];
            a2.x = ap[32 * KC];     a2.y = ap[32 * KC + 1];
            a3.x = ap[48 * KC];     a3.y = ap[48 * KC + 1];
            acc0 = __builtin_amdgcn_wmma_f32_16x16x4_f32(false, a0, false, b, (short)0, acc0, false, false);
            acc1 = __builtin_amdgcn_wmma_f32_16x16x4_f32(false, a1, false, b, (short)0, acc1, false, false);
            acc2 = __builtin_amdgcn_wmma_f32_16x16x4_f32(false, a2, false, b, (short)0, acc2, false, false);
            acc3 = __builtin_amdgcn_wmma_f32_16x16x4_f32(false, a3, false, b, (short)0, acc3, false, false);
        }
    }

    // Spill accumulator tiles to LDS: VGPR j holds row (j + 8*h), cols n0+m
    {
        float* dst = (wv < 4) ? cleanL : sdL;
        #pragma unroll
        for (int j = 0; j < 8; ++j) {
            dst[( 0 + j + 8 * h) * 64 + n0 + m] = acc0[j];
            dst[(16 + j + 8 * h) * 64 + n0 + m] = acc1[j];
            dst[(32 + j + 8 * h) * 64 + n0 + m] = acc2[j];
            dst[(48 + j + 8 * h) * 64 + n0 + m] = acc3[j];
        }
    }
    __syncthreads();

    // Phase 1: stddev = softplus(noise_logit) + eps; noisy = clean + noise*stddev
    for (int i = tid; i < M_TILE * 64; i += 256) {
        const int r = i >> 6;
        const int e = i & 63;
        const float v  = sdL[i];
        const float sp = (v > 20.0f) ? v : log1pf(__expf(v));
        const float sd = sp + 0.01f;
        sdL[i]    = sd;
        noisyL[i] = cleanL[i] + noise[(size_t)(row0 + r) * E_DIM + e] * sd;
    }
    __syncthreads();

    // Phase 2: per-row softmax, top-9, top-8 gate softmax, load contribution
    if (tid < M_TILE) {
        const int r = tid;
        float* noisyRow = &noisyL[r * 64];
        float* probsRow = &probsL[r * 64];
        float* cleanRow = &cleanL[r * 64];
        float* sdRow    = &sdL[r * 64];

        // softmax over 64 experts
        float mx = -3.4e38f;
        for (int e = 0; e < 64; ++e) mx = fmaxf(mx, noisyRow[e]);
        float s = 0.0f;
        for (int e = 0; e < 64; ++e) {
            const float ev = __expf(noisyRow[e] - mx);
            probsRow[e] = ev;
            s += ev;
        }
        const float inv = 1.0f / s;
        for (int e = 0; e < 64; ++e) probsRow[e] *= inv;

        // top-(TOPK+1) by repeated argmax
        unsigned long long used = 0ull;
        float topv[TOPK_N + 1];
        int   topIdx[TOPK_N];
        for (int k = 0; k <= TOPK_N; ++k) {
            float best = -1.0f;
            int   bi   = 0;
            for (int e = 0; e < 64; ++e) {
                const bool avail = ((used >> e) & 1ull) == 0ull;
                const float v = probsRow[e];
                if (avail && v > best) { best = v; bi = e; }
            }
            used |= (1ull << bi);
            topv[k] = best;
            if (k < TOPK_N) topIdx[k] = bi;
        }

        // softmax over the top-8 prob values (reference semantics)
        const float m0 = topv[0];
        float gv[TOPK_N];
        float gs = 0.0f;
        #pragma unroll
        for (int k = 0; k < TOPK_N; ++k) { gv[k] = __expf(topv[k] - m0); gs += gv[k]; }
        const float ginv = 1.0f / gs;

        // reuse probsRow as the gates row (zero + scatter)
        for (int e = 0; e < 64; ++e) probsRow[e] = 0.0f;
        #pragma unroll
        for (int k = 0; k < TOPK_N; ++k) probsRow[topIdx[k]] = gv[k] * ginv;

        // load contribution: normcdf((clean - thr)/stddev); overwrite cleanL
        const float thrIn  = topv[TOPK_N];
        const float thrOut = topv[TOPK_N - 1];
        for (int e = 0; e < 64; ++e) {
            const float th  = (noisyRow[e] > thrIn) ? thrIn : thrOut;
            const float arg = (cleanRow[e] - th) / sdRow[e];
            cleanRow[e] = 0.5f * (1.0f + erff(arg * 0.70710678118654752f));
        }
    }
    __syncthreads();

    // Coalesced gates store: 4 float4s per thread (same mapping as staging)
    {
        #pragma unroll
        for (int t = 0; t < 4; ++t) {
            const int r = sr + 16 * t;
            const float4 g = *(const float4*)(&probsL[r * 64 + sc]);
            *(float4*)(gates_out + (size_t)(row0 + r) * 64 + sc) = g;
        }
    }

    // Column-reduce load contributions (stored in cleanL), one atomic per expert
    if (tid < 64) {
        float colsum = 0.0f;
        for (int r = 0; r < M_TILE; ++r) colsum += cleanL[r * 64 + tid];
        atomicAdd(&load_out[tid], colsum);
    }
}

extern "C" void kernel_launch(void* const* d_in, const int* in_sizes, int n_in,
                              void* d_out, int out_size, void* d_ws, size_t ws_size,
                              hipStream_t stream) {
    const float* x       = (const float*)d_in[0];
    const float* w_gate  = (const float*)d_in[1];
    const float* w_noise = (const float*)d_in[2];
    const float* noise   = (const float*)d_in[3];

    float* gates = (float*)d_out;                       // B_ROWS * E_DIM floats
    float* load  = gates + (size_t)B_ROWS * E_DIM;      // E_DIM floats

    moe_zero_load_kernel<<<1, 64, 0, stream>>>(load);
    moe_router_kernel<<<B_ROWS / M_TILE, 256, 0, stream>>>(
        x, w_gate, w_noise, noise, gates, load);
}